// PointSetCls_75986561401147
// MI455X (gfx1250) — compile-verified
//
#include <hip/hip_runtime.h>
#include <math.h>

typedef __attribute__((ext_vector_type(2))) float v2f;
typedef __attribute__((ext_vector_type(8))) float v8f;

#define BATCH 32
#define NPTS  8192

// ---- workspace layout (in 32-bit words) ----
#define WS_MX1    0        // 96   u32  flipped max_n x               per (b,3)
#define WS_MXP2   128      // 2048 u32  flipped max_n pre1            per (b,64)
#define WS_MXA2   2176     // 4096 u32  flipped max_n acc2 (pre-bias) per (b,128)
#define WS_GMAX   6272     // 8192 u32  flipped max_n acc3 (pre-bias) per (b,256)
#define WS_C1     14464    // 2048 f32
#define WS_C2     16512    // 4096 f32
#define WS_C3     20608    // 8192 f32
#define WS_H2     28800    // 32*128*8192 f32 = 128 MiB
#define WS_INIT_WORDS 14464

__device__ __forceinline__ unsigned flipf(float f) {
    unsigned u = __float_as_uint(f);
    return (u & 0x80000000u) ? ~u : (u | 0x80000000u);
}
__device__ __forceinline__ float unflipf(unsigned u) {
    u = (u & 0x80000000u) ? (u & 0x7fffffffu) : ~u;
    return __uint_as_float(u);
}
// branch-free tanh: v_exp_f32 + v_rcp_f32 (TRANS ops, co-issue with WMMA)
__device__ __forceinline__ float fast_tanh(float x) {
    float t = __builtin_amdgcn_exp2f(x * 2.8853900817779268f);  // exp(2x)
    return 1.0f - 2.0f * __builtin_amdgcn_rcpf(t + 1.0f);
}

// ---------------- K0: reset flipped-max accumulators ----------------
__global__ void k_init(unsigned* __restrict__ ws) {
    int i = blockIdx.x * 256 + threadIdx.x;
    if (i < WS_INIT_WORDS) ws[i] = 0u;
}

// ---------------- K1: mx1 = max_n x[b][f][n] ----------------
__global__ void k_xmax(const float* __restrict__ x, unsigned* __restrict__ mx1) {
    int b = blockIdx.y;
    int col = blockIdx.x * 256 + threadIdx.x;
    __shared__ unsigned s[3];
    if (threadIdx.x < 3) s[threadIdx.x] = 0u;
    __syncthreads();
    #pragma unroll
    for (int f = 0; f < 3; ++f)
        atomicMax(&s[f], flipf(x[((size_t)b * 3 + f) * NPTS + col]));
    __syncthreads();
    if (threadIdx.x < 3) atomicMax(&mx1[b * 3 + threadIdx.x], s[threadIdx.x]);
}

// --- adjusted bias: c[b][o] = bias[o] - sum_f w[o][f] * act(unflip(mx[b][f]) + add[b][f])
__global__ void k_cvec(const float* __restrict__ w, const float* __restrict__ bias,
                       const unsigned* __restrict__ mxf, const float* __restrict__ add,
                       float* __restrict__ c, int O, int F, int apply_tanh) {
    int idx = blockIdx.x * 256 + threadIdx.x;
    if (idx >= BATCH * O) return;
    int b = idx / O, o = idx % O;
    float s = bias[o];
    for (int f = 0; f < F; ++f) {
        float m = unflipf(mxf[b * F + f]);
        if (add) m += add[b * F + f];
        if (apply_tanh) m = tanhf(m);
        s = fmaf(-w[o * F + f], m, s);
    }
    c[idx] = s;
}

// ---------------- K2: mxp2 = max_n (W1 x + c1)   (pre-activation; tanh deferred) ----------
__global__ void k_pre1max(const float* __restrict__ x, const float* __restrict__ w1,
                          const float* __restrict__ c1, unsigned* __restrict__ mxp2) {
    int b = blockIdx.y;
    int col = blockIdx.x * 256 + threadIdx.x;
    __shared__ unsigned s[64];
    if (threadIdx.x < 64) s[threadIdx.x] = 0u;
    __syncthreads();
    float x0 = x[((size_t)b * 3 + 0) * NPTS + col];
    float x1 = x[((size_t)b * 3 + 1) * NPTS + col];
    float x2 = x[((size_t)b * 3 + 2) * NPTS + col];
    for (int o = 0; o < 64; ++o) {
        float p = fmaf(w1[o * 3 + 0], x0,
                  fmaf(w1[o * 3 + 1], x1,
                  fmaf(w1[o * 3 + 2], x2, c1[b * 64 + o])));
        atomicMax(&s[o], flipf(p));
    }
    __syncthreads();
    if (threadIdx.x < 64) atomicMax(&mxp2[b * 64 + threadIdx.x], s[threadIdx.x]);
}

// ---------------- K3: layer2 WMMA ----------------------------------------------------
// block 256 = 8 waves; wave w owns M-tile w (W2 fragments resident in VGPRs).
// Block stages h1 panel (computed on the fly) in LDS transposed [col][k].
// grid (NPTS/128, BATCH)
#define L2_STRIDE 68   // 64 k + pad (stride%64banks = 4 -> conflict-light)
__global__ void k_layer2(const float* __restrict__ x, const float* __restrict__ w1,
                         const float* __restrict__ c1, const float* __restrict__ w2,
                         const float* __restrict__ c2, float* __restrict__ h2,
                         unsigned* __restrict__ mxa2) {
    const int b = blockIdx.y;
    const int colbase = blockIdx.x * 128;
    const int wave = threadIdx.x >> 5;        // == M-tile
    const int lane = threadIdx.x & 31;
    const int half = lane >> 4;
    const int n    = lane & 15;

    __shared__ float h1s[128 * L2_STRIDE];

    // stage h1 = fast_tanh(W1 x + c1), transposed [col][k]
    {
        int col = threadIdx.x & 127;
        int kb  = (threadIdx.x >> 7) * 32;    // {0,32}
        int cg  = colbase + col;
        float x0 = x[((size_t)b * 3 + 0) * NPTS + cg];
        float x1 = x[((size_t)b * 3 + 1) * NPTS + cg];
        float x2 = x[((size_t)b * 3 + 2) * NPTS + cg];
        #pragma unroll
        for (int k = 0; k < 32; ++k) {
            int o = kb + k;
            float p = fmaf(w1[o * 3 + 0], x0,
                      fmaf(w1[o * 3 + 1], x1,
                      fmaf(w1[o * 3 + 2], x2, c1[b * 64 + o])));
            h1s[col * L2_STRIDE + o] = fast_tanh(p);
        }
    }
    __syncthreads();

    // resident A fragments (W2 rows for this wave's M-tile)
    v2f afr[16];
    #pragma unroll
    for (int kk = 0; kk < 16; ++kk) {
        int k0 = kk * 4 + 2 * half;
        const float* wr = &w2[(size_t)(wave * 16 + n) * 64 + k0];
        v2f a; a.x = wr[0]; a.y = wr[1];
        afr[kk] = a;
    }
    float c2v[8];
    #pragma unroll
    for (int g = 0; g < 8; ++g) c2v[g] = c2[b * 128 + wave * 16 + g + 8 * half];

    float vmax[8];
    #pragma unroll
    for (int g = 0; g < 8; ++g) vmax[g] = -3.4e38f;

    for (int ct = 0; ct < 8; ++ct) {
        int col = ct * 16 + n;
        v8f acc = {};
        #pragma unroll
        for (int kk = 0; kk < 16; ++kk) {
            int k0 = kk * 4 + 2 * half;
            v2f bf = *(const v2f*)&h1s[col * L2_STRIDE + k0];
            acc = __builtin_amdgcn_wmma_f32_16x16x4_f32(
                false, afr[kk], false, bf, (short)0, acc, false, false);
        }
        #pragma unroll
        for (int g = 0; g < 8; ++g) {
            int o = wave * 16 + g + 8 * half;
            vmax[g] = fmaxf(vmax[g], acc[g]);
            h2[((size_t)b * 128 + o) * NPTS + colbase + col] = fast_tanh(acc[g] + c2v[g]);
        }
    }
    // cross-lane max over the 16 columns of each half-wave
    #pragma unroll
    for (int g = 0; g < 8; ++g) {
        #pragma unroll
        for (int m = 1; m < 16; m <<= 1)
            vmax[g] = fmaxf(vmax[g], __shfl_xor(vmax[g], m, 32));
    }
    if (n == 0) {
        #pragma unroll
        for (int g = 0; g < 8; ++g)
            atomicMax(&mxa2[b * 128 + wave * 16 + g + 8 * half], flipf(vmax[g]));
    }
}

// ---------------- K4: layer3 WMMA + fused global max pool (no h3, no tanh) -----------
// block 512 = 16 waves; wave w owns M-tile w (W3 fragments resident: 64 VGPRs).
// h2 panel staged in LDS transposed [col][k] (128x132 f32 = 67.5 KB of 320 KB WGP LDS).
// grid (NPTS/128, BATCH)
#define L3_STRIDE 132
__global__ void k_layer3(const float* __restrict__ h2, const float* __restrict__ w3,
                         unsigned* __restrict__ gmax) {
    const int b = blockIdx.y;
    const int colbase = blockIdx.x * 128;
    const int wave = threadIdx.x >> 5;        // == M-tile (0..15)
    const int lane = threadIdx.x & 31;
    const int half = lane >> 4;
    const int n    = lane & 15;

    __shared__ float h2s[128 * L3_STRIDE];

    // stage h2 panel, transposed [col][k]
    {
        int col = threadIdx.x & 127;
        int kb  = (threadIdx.x >> 7) * 32;    // {0,32,64,96}
        const float* src = &h2[((size_t)b * 128 + kb) * NPTS + colbase + col];
        #pragma unroll
        for (int k = 0; k < 32; ++k)
            h2s[col * L3_STRIDE + kb + k] = src[(size_t)k * NPTS];
    }
    __syncthreads();

    // resident A fragments (W3 rows for this wave's M-tile)
    v2f afr[32];
    #pragma unroll
    for (int kk = 0; kk < 32; ++kk) {
        int k0 = kk * 4 + 2 * half;
        const float* wr = &w3[(size_t)(wave * 16 + n) * 128 + k0];
        v2f a; a.x = wr[0]; a.y = wr[1];
        afr[kk] = a;
    }

    float vmax[8];
    #pragma unroll
    for (int g = 0; g < 8; ++g) vmax[g] = -3.4e38f;

    for (int ct = 0; ct < 8; ++ct) {
        int col = ct * 16 + n;
        v8f acc = {};
        #pragma unroll
        for (int kk = 0; kk < 32; ++kk) {
            int k0 = kk * 4 + 2 * half;
            v2f bf = *(const v2f*)&h2s[col * L3_STRIDE + k0];
            acc = __builtin_amdgcn_wmma_f32_16x16x4_f32(
                false, afr[kk], false, bf, (short)0, acc, false, false);
        }
        #pragma unroll
        for (int g = 0; g < 8; ++g) vmax[g] = fmaxf(vmax[g], acc[g]);
    }
    #pragma unroll
    for (int g = 0; g < 8; ++g) {
        #pragma unroll
        for (int m = 1; m < 16; m <<= 1)
            vmax[g] = fmaxf(vmax[g], __shfl_xor(vmax[g], m, 32));
    }
    if (n == 0) {
        #pragma unroll
        for (int g = 0; g < 8; ++g)
            atomicMax(&gmax[b * 256 + wave * 16 + g + 8 * half], flipf(vmax[g]));
    }
}

// ---------------- K5: FC head + log_softmax (one block per sample) ----------------
__global__ void k_head(const unsigned* __restrict__ gmax, const float* __restrict__ c3,
                       const float* __restrict__ fc1w, const float* __restrict__ fc1b,
                       const float* __restrict__ fc2w, const float* __restrict__ fc2b,
                       float* __restrict__ out) {
    int b = blockIdx.x, t = threadIdx.x;
    __shared__ float gsh[256], a1[128], zsh[40], red[2];
    gsh[t] = tanhf(unflipf(gmax[b * 256 + t]) + c3[b * 256 + t]);
    __syncthreads();
    if (t < 128) {
        float s = fc1b[t];
        for (int f = 0; f < 256; ++f) s = fmaf(fc1w[t * 256 + f], gsh[f], s);
        a1[t] = tanhf(s);
    }
    __syncthreads();
    if (t < 40) {
        float s = fc2b[t];
        for (int f = 0; f < 128; ++f) s = fmaf(fc2w[t * 128 + f], a1[f], s);
        zsh[t] = tanhf(s);
    }
    __syncthreads();
    if (t == 0) {
        float m = -1e30f;
        for (int i = 0; i < 40; ++i) m = fmaxf(m, zsh[i]);
        float se = 0.f;
        for (int i = 0; i < 40; ++i) se += expf(zsh[i] - m);
        red[0] = m; red[1] = logf(se);
    }
    __syncthreads();
    if (t < 40) out[b * 40 + t] = zsh[t] - red[0] - red[1];
}

extern "C" void kernel_launch(void* const* d_in, const int* in_sizes, int n_in,
                              void* d_out, int out_size, void* d_ws, size_t ws_size,
                              hipStream_t stream) {
    const float* x    = (const float*)d_in[0];
    const float* w1   = (const float*)d_in[1];
    const float* b1   = (const float*)d_in[2];
    const float* w2   = (const float*)d_in[3];
    const float* b2   = (const float*)d_in[4];
    const float* w3   = (const float*)d_in[5];
    const float* b3   = (const float*)d_in[6];
    const float* fc1w = (const float*)d_in[7];
    const float* fc1b = (const float*)d_in[8];
    const float* fc2w = (const float*)d_in[9];
    const float* fc2b = (const float*)d_in[10];
    float* out = (float*)d_out;

    unsigned* wsu = (unsigned*)d_ws;
    float*    wsf = (float*)d_ws;
    unsigned* mx1  = wsu + WS_MX1;
    unsigned* mxp2 = wsu + WS_MXP2;
    unsigned* mxa2 = wsu + WS_MXA2;
    unsigned* gmax = wsu + WS_GMAX;
    float*    c1   = wsf + WS_C1;
    float*    c2   = wsf + WS_C2;
    float*    c3   = wsf + WS_C3;
    float*    h2   = wsf + WS_H2;

    k_init<<<(WS_INIT_WORDS + 255) / 256, 256, 0, stream>>>(wsu);
    k_xmax<<<dim3(NPTS / 256, BATCH), 256, 0, stream>>>(x, mx1);
    // c1 = b1 - W1 . mx1
    k_cvec<<<(BATCH * 64 + 255) / 256, 256, 0, stream>>>(w1, b1, mx1, nullptr, c1, 64, 3, 0);
    k_pre1max<<<dim3(NPTS / 256, BATCH), 256, 0, stream>>>(x, w1, c1, mxp2);
    // c2 = b2 - W2 . tanh(max pre1)
    k_cvec<<<(BATCH * 128 + 255) / 256, 256, 0, stream>>>(w2, b2, mxp2, nullptr, c2, 128, 64, 1);
    k_layer2<<<dim3(NPTS / 128, BATCH), 256, 0, stream>>>(x, w1, c1, w2, c2, h2, mxa2);
    // c3 = b3 - W3 . tanh(max acc2 + c2)
    k_cvec<<<(BATCH * 256 + 255) / 256, 256, 0, stream>>>(w3, b3, mxa2, c2, c3, 256, 128, 1);
    k_layer3<<<dim3(NPTS / 128, BATCH), 512, 0, stream>>>(h2, w3, gmax);
    k_head<<<BATCH, 256, 0, stream>>>(gmax, c3, fc1w, fc1b, fc2w, fc2b, out);
}